// UMA_21852793602400
// MI455X (gfx1250) — compile-verified
//
#include <hip/hip_runtime.h>
#include <math.h>

typedef __attribute__((ext_vector_type(2))) float v2f;
typedef __attribute__((ext_vector_type(8))) float v8f;

#define Bc 32
#define Lc 8192
#define Dc 256
#define Mc 4097   // L/2 + 1

// ---------------------------------------------------------------------------
// Kernel 1: s[row] = sigmoid(dot(x[row], W) + b) via V_WMMA_F32_16X16X4_F32.
// One wave handles 16 rows. B-matrix columns all hold the same W chunk, so
// every column of D replicates the 16 dot products. 64 chained WMMAs (K=256).
// A layout (ISA 7.12.2, 32-bit A 16x4): lane<16 -> row=lane, K={k,k+1};
// lane>=16 -> row=lane-16, K={k+2,k+3}. B mirrors this with W values.
// D layout: VGPR j = row j (lanes 0-15) / row j+8 (lanes 16-31).
// ---------------------------------------------------------------------------
__global__ __launch_bounds__(256) void uma_score_wmma(
    const float* __restrict__ x, const float* __restrict__ W,
    const float* __restrict__ bias, float* __restrict__ s_out) {
  __shared__ __align__(16) float Wl[Dc];
  int t = threadIdx.x;
  Wl[t] = W[t];
  __syncthreads();

  int lane = t & 31;
  int wave = t >> 5;
  int tile = blockIdx.x * 8 + wave;     // exact tiling: 2048 blocks * 8 waves
  int rowBase = tile * 16;
  int mrel = lane & 15;
  int koff = (lane >> 4) * 2;           // 0 for lanes 0-15, 2 for lanes 16-31

  const float* xr = x + ((size_t)(rowBase + mrel) << 8);  // 256 floats/row
  v8f c = {};
#pragma unroll
  for (int k = 0; k < Dc; k += 4) {
    int kk = k + koff;
    v2f a = *(const v2f*)(xr + kk);       // global_load_b64 (A operand)
    v2f bm = *(const v2f*)(&Wl[kk]);      // ds_load_b64     (B operand)
    c = __builtin_amdgcn_wmma_f32_16x16x4_f32(
        /*neg_a=*/false, a, /*neg_b=*/false, bm,
        /*c_mod=*/(short)0, c, /*reuse_a=*/false, /*reuse_b=*/false);
  }

  float b0 = bias[0];
  if (mrel == 0) {                      // lane 0 -> rows 0..7, lane 16 -> 8..15
    int base = rowBase + koff * 4;
#pragma unroll
    for (int j = 0; j < 8; ++j) {
      float v = c[j] + b0;
      s_out[base + j] = 1.0f / (1.0f + expf(-v));
    }
  }
}

// ---------------------------------------------------------------------------
// Kernel 2: strict local minima -> ordered compaction per batch.
// One block per batch; 256 threads x 32 contiguous positions each; block-wide
// exclusive scan in LDS keeps the compaction in ascending-position order
// (equivalent to sort(where(mask, pos, L))[:M]).
// ---------------------------------------------------------------------------
__global__ __launch_bounds__(256) void uma_mask_scan(
    const float* __restrict__ s, int* __restrict__ starts,
    int* __restrict__ counts, int* __restrict__ counts_out) {
  __shared__ int sc[256];
  int b = blockIdx.x;
  int t = threadIdx.x;
  const float* sb = s + (size_t)b * Lc;
  int base = t * 32;

  int cnt = 0;
  for (int j = 0; j < 32; ++j) {
    int i = base + j;
    float si = sb[i];
    float sp = (i > 0) ? sb[i - 1] : 0.0f;        // zero padding at edges
    float sn = (i < Lc - 1) ? sb[i + 1] : 0.0f;
    bool m = (i == 0) || ((si < sp) && (si < sn)); // mask[0] forced True
    cnt += m ? 1 : 0;
  }

  sc[t] = cnt;
  __syncthreads();
  for (int off = 1; off < 256; off <<= 1) {
    int v = sc[t];
    int add = (t >= off) ? sc[t - off] : 0;
    __syncthreads();
    sc[t] = v + add;
    __syncthreads();
  }
  int excl = sc[t] - cnt;    // exclusive prefix
  int total = sc[255];

  int* stb = starts + (size_t)b * Mc;
  int o = excl;
  for (int j = 0; j < 32; ++j) {
    int i = base + j;
    float si = sb[i];
    float sp = (i > 0) ? sb[i - 1] : 0.0f;
    float sn = (i < Lc - 1) ? sb[i + 1] : 0.0f;
    bool m = (i == 0) || ((si < sp) && (si < sn));
    if (m) stb[o++] = i;
  }
  if (t == 0) { counts[b] = total; counts_out[b] = total; }
}

// ---------------------------------------------------------------------------
// Kernel 3: one wave per output row [b][m]. Valid rows gather x[start] and
// x[start+1] (clipped/weighted), scale by s, normalize by max(den,1e-6).
// Invalid rows (m >= count) are written as exact zeros.
// ---------------------------------------------------------------------------
__global__ __launch_bounds__(256) void uma_gather(
    const float* __restrict__ x, const float* __restrict__ s,
    const int* __restrict__ starts, const int* __restrict__ counts,
    float* __restrict__ out) {
  int t = threadIdx.x;
  int lane = t & 31;
  int wave = t >> 5;
  long long r = (long long)blockIdx.x * 8 + wave;
  if (r >= (long long)Bc * Mc) return;
  int b = (int)(r / Mc);
  int m = (int)(r % Mc);

  float4* dst = (float4*)(out + (size_t)r * Dc);
  int cnt = counts[b];
  if (m < cnt) {
    int st = starts[(size_t)b * Mc + m];
    int i1 = (st + 1 < Lc) ? st + 1 : Lc - 1;
    float w1 = (st + 1 < Lc) ? 1.0f : 0.0f;
    float s0 = s[(size_t)b * Lc + st];
    float s1 = s[(size_t)b * Lc + i1] * w1;
    float inv = 1.0f / fmaxf(s0 + s1, 1e-6f);
    const float4* x0 = (const float4*)(x + ((size_t)b * Lc + st) * Dc);
    const float4* x1 = (const float4*)(x + ((size_t)b * Lc + i1) * Dc);
#pragma unroll
    for (int j = 0; j < 2; ++j) {
      int idx = lane * 2 + j;           // 8 floats per lane, coalesced
      float4 a = x0[idx];
      float4 c2 = x1[idx];
      float4 rr;
      rr.x = (s0 * a.x + s1 * c2.x) * inv;
      rr.y = (s0 * a.y + s1 * c2.y) * inv;
      rr.z = (s0 * a.z + s1 * c2.z) * inv;
      rr.w = (s0 * a.w + s1 * c2.w) * inv;
      dst[idx] = rr;
    }
  } else {
    float4 z = make_float4(0.0f, 0.0f, 0.0f, 0.0f);
    dst[lane * 2] = z;
    dst[lane * 2 + 1] = z;
  }
}

// ---------------------------------------------------------------------------
extern "C" void kernel_launch(void* const* d_in, const int* in_sizes, int n_in,
                              void* d_out, int out_size, void* d_ws, size_t ws_size,
                              hipStream_t stream) {
  const float* x    = (const float*)d_in[0];   // xs_pad [32,8192,256] f32
  // d_in[1] = olens (unused by the reference computation)
  const float* W    = (const float*)d_in[2];   // [256,1] f32
  const float* bias = (const float*)d_in[3];   // [1] f32

  float* out = (float*)d_out;                                  // [32,4097,256]
  int* counts_out = (int*)out + (size_t)Bc * Mc * Dc;          // 32 ints (tail)

  float* s    = (float*)d_ws;                                  // [32,8192] f32
  int*  starts = (int*)((char*)d_ws + (size_t)Bc * Lc * sizeof(float)); // [32,4097]
  int*  counts = starts + (size_t)Bc * Mc;                     // [32]

  // K1: 262144 rows / 16 per wave / 8 waves per block = 2048 blocks.
  uma_score_wmma<<<(Bc * Lc / 16) / 8, 256, 0, stream>>>(x, W, bias, s);
  // K2: one block per batch.
  uma_mask_scan<<<Bc, 256, 0, stream>>>(s, starts, counts, counts_out);
  // K3: 131104 output rows, 8 waves per block.
  uma_gather<<<(Bc * Mc + 7) / 8, 256, 0, stream>>>(x, s, starts, counts, out);
}